// CL_88399016886706
// MI455X (gfx1250) — compile-verified
//
#include <hip/hip_runtime.h>
#include <hip/hip_bf16.h>
#include <math.h>

#define K_TOPICS   8192
#define EMB        1024
#define N_CLUST    192
#define WL_CAP     16384

typedef __attribute__((ext_vector_type(2))) float v2f;
typedef __attribute__((ext_vector_type(8))) float v8f;

// ---------------- kernel 1: inverse norms ----------------
__global__ __launch_bounds__(256) void norms_kernel(const float* __restrict__ x,
                                                    float* __restrict__ invnorm) {
    __shared__ float red[256];
    const int r = blockIdx.x;
    const float* row = x + (size_t)r * EMB;
    float s = 0.f;
    for (int k = threadIdx.x; k < EMB; k += 256) { float v = row[k]; s += v * v; }
    red[threadIdx.x] = s;
    __syncthreads();
    for (int off = 128; off > 0; off >>= 1) {
        if (threadIdx.x < off) red[threadIdx.x] += red[threadIdx.x + off];
        __syncthreads();
    }
    if (threadIdx.x == 0) invnorm[r] = 1.0f / sqrtf(red[0]);
}

// ---------------- kernel 2: bucket by cluster + build tile worklist ----------------
__global__ __launch_bounds__(256) void build_kernel(const int* __restrict__ cid,
                                                    int* counts, int* offsets, int* cursor,
                                                    int* members, int* workcount,
                                                    int* worklist, float* out) {
    const int tid = threadIdx.x;
    for (int c = tid; c < N_CLUST; c += 256) { counts[c] = 0; cursor[c] = 0; }
    if (tid == 0) { *workcount = 0; *out = 0.f; }
    __syncthreads();
    for (int t = tid; t < K_TOPICS; t += 256) atomicAdd(&counts[cid[t]], 1);
    __syncthreads();
    if (tid == 0) {
        int off = 0;
        for (int c = 0; c < N_CLUST; ++c) { offsets[c] = off; off += counts[c]; }
    }
    __syncthreads();
    for (int t = tid; t < K_TOPICS; t += 256) {
        int c = cid[t];
        int p = atomicAdd(&cursor[c], 1);
        members[offsets[c] + p] = t;
    }
    __syncthreads();
    for (int c = tid; c < N_CLUST; c += 256) {
        int n = counts[c];
        int T = (n + 15) >> 4;
        int nit = T * (T + 1) / 2;
        if (nit == 0) continue;
        int base = atomicAdd(workcount, nit);
        int k = 0;
        for (int ti = 0; ti < T; ++ti)
            for (int tj = ti; tj < T; ++tj, ++k) {
                int idx = base + k;
                if (idx < WL_CAP) worklist[idx] = c | (ti << 8) | (tj << 20);
            }
    }
}

// ---------------- kernel 3: per-tile Gram via fp32 WMMA + hinge reduction ----------------
__global__ __launch_bounds__(256) void gram_kernel(const float* __restrict__ x,
                                                   const float* __restrict__ invnorm,
                                                   const int* __restrict__ counts,
                                                   const int* __restrict__ offsets,
                                                   const int* __restrict__ members,
                                                   const int* __restrict__ workcount,
                                                   const int* __restrict__ worklist,
                                                   float* __restrict__ out) {
    const int lane = threadIdx.x & 31;
    const int lo   = lane & 15;      // = M (A frag) / N (B frag, C col)
    const int hi   = lane >> 4;      // K-half selector; C row half
    const int wavesPerBlock = blockDim.x >> 5;
    const int waveId = blockIdx.x * wavesPerBlock + (threadIdx.x >> 5);
    const int nWaves = gridDim.x * wavesPerBlock;

    int nitems = *workcount;
    if (nitems > WL_CAP) nitems = WL_CAP;

    float lsum = 0.f;
    for (int it = waveId; it < nitems; it += nWaves) {
        const int w  = worklist[it];
        const int c  = w & 255;
        const int ti = (w >> 8) & 0xFFF;
        const int tj = (w >> 20) & 0xFFF;
        const int n   = counts[c];
        const int off = offsets[c];

        const int miA = ti * 16 + lo;
        const int miB = tj * 16 + lo;
        const int validA = miA < n;
        const int validB = miB < n;
        const int rowA = members[off + (validA ? miA : 0)];
        const int rowB = members[off + (validB ? miB : 0)];
        const float invA = invnorm[rowA];
        const float invB = invnorm[rowB];
        const float fvA  = validA ? 1.0f : 0.0f;
        const float fvB  = validB ? 1.0f : 0.0f;

        // fp32 WMMA fragment addressing: lane holds M/N = lo, K = 2*hi + {0,1}
        const v2f* pa = (const v2f*)(x + (size_t)rowA * EMB);
        const v2f* pb = (const v2f*)(x + (size_t)rowB * EMB);

        v8f acc0 = {};
        v8f acc1 = {};
#pragma unroll 4
        for (int t = 0; t < 128; ++t) {            // 2 x (K=4) per iter -> K=1024
            v2f a0 = pa[4 * t + hi];
            v2f b0 = pb[4 * t + hi];
            v2f a1 = pa[4 * t + 2 + hi];
            v2f b1 = pb[4 * t + 2 + hi];
            acc0 = __builtin_amdgcn_wmma_f32_16x16x4_f32(
                false, a0, false, b0, (short)0, acc0, false, false);
            acc1 = __builtin_amdgcn_wmma_f32_16x16x4_f32(
                false, a1, false, b1, (short)0, acc1, false, false);
        }

        // C/D layout: VGPR v of lane -> M = v + 8*hi, N = lo
        const int diag = (ti == tj);
#pragma unroll
        for (int v = 0; v < 8; ++v) {
            const int m = v + 8 * hi;
            float d  = acc0[v] + acc1[v];
            float iA = __shfl(invA, m, 32);   // invnorm of tile-I row m (held by lane m)
            float mA = __shfl(fvA,  m, 32);   // validity of tile-I row m
            float sim = d * iA * invB;
            float contrib = (sim > 0.5f) ? fmaxf(1.0f - sim, 0.0f) : fmaxf(sim, 0.0f);
            float keep = mA * fvB;
            if (diag && m >= lo) keep = 0.0f; // strict upper triangle on diagonal tiles
            lsum += keep * contrib;
        }
    }

    // wave32 reduction, one global atomic per wave
    for (int s = 16; s > 0; s >>= 1) lsum += __shfl_xor(lsum, s, 32);
    if (lane == 0) atomicAdd(out, lsum);
}

// ---------------- host entry ----------------
extern "C" void kernel_launch(void* const* d_in, const int* in_sizes, int n_in,
                              void* d_out, int out_size, void* d_ws, size_t ws_size,
                              hipStream_t stream) {
    const float* x   = (const float*)d_in[0];
    const int*   cid = (const int*)d_in[1];
    float* out = (float*)d_out;

    // ws layout (4-byte units)
    float* invnorm  = (float*)d_ws;             // 8192
    int* counts     = (int*)d_ws + K_TOPICS;    // 192
    int* offsets    = counts + N_CLUST;         // 192
    int* cursor     = offsets + N_CLUST;        // 192
    int* members    = cursor + N_CLUST;         // 8192
    int* workcount  = members + K_TOPICS;       // 1
    int* worklist   = workcount + 1;            // WL_CAP

    norms_kernel<<<K_TOPICS, 256, 0, stream>>>(x, invnorm);
    build_kernel<<<1, 256, 0, stream>>>(cid, counts, offsets, cursor,
                                        members, workcount, worklist, out);
    gram_kernel<<<256, 256, 0, stream>>>(x, invnorm, counts, offsets,
                                         members, workcount, worklist, out);
}